// AgeUGP_v2_30030411334318
// MI455X (gfx1250) — compile-verified
//
#include <hip/hip_runtime.h>
#include <hip/hip_bf16.h>
#include <math.h>

#define B_DIM    8
#define N_SNPS   500000
#define N_NODES  2000000
#define N_GENES  20000
#define N_FILT   8
#define H1       1024
#define H2       256
#define H3       16

#define G1_NTILES   (H1 / 16)      // 64
#define G1_KSPLIT   25             // 25 K-chunks of 800; per wave 100 K = 25 wmma
#define G1_KCHUNK   (N_GENES / G1_KSPLIT)   // 800
#define G1_KWAVE    (G1_KCHUNK / 8)         // 100

typedef __attribute__((ext_vector_type(2))) float v2f;
typedef __attribute__((ext_vector_type(8))) float v8f;

// ---------------------------------------------------------------------------
// Kernel 1: fsum[n] = sum_f filters[f][n]   (N_FILT axis of the einsum
// factors out of the downstream mean over f)
// ---------------------------------------------------------------------------
__global__ void k_fsum(const float* __restrict__ filters, float* __restrict__ fsum) {
    int n = blockIdx.x * blockDim.x + threadIdx.x;
    if (n < N_SNPS) {
        float s = 0.f;
#pragma unroll
        for (int f = 0; f < N_FILT; ++f) s += filters[f * N_SNPS + n];
        fsum[n] = s;
    }
}

// ---------------------------------------------------------------------------
// Kernel 2: one wave32 per gene. node_seg is sorted -> binary-search bounds.
// Lanes strip-mine the node range; butterfly-reduce 8 batch accumulators.
// Writes A padded to 16 rows (rows 8..15 = 0), row-major (16 x N_GENES).
// ---------------------------------------------------------------------------
__global__ void k_segsum(const float* __restrict__ snp,
                         const int*   __restrict__ snp_ids,
                         const int*   __restrict__ node_seg,
                         const float* __restrict__ fsum,
                         float*       __restrict__ Apad) {
    int wave = (blockIdx.x * blockDim.x + threadIdx.x) >> 5;
    int lane = threadIdx.x & 31;
    if (wave >= N_GENES) return;
    int g = wave;

    int s_lo = 0, s_hi = N_NODES;
    while (s_lo < s_hi) { int mid = (s_lo + s_hi) >> 1; if (node_seg[mid] <  g) s_lo = mid + 1; else s_hi = mid; }
    int e_lo = s_lo, e_hi = N_NODES;
    while (e_lo < e_hi) { int mid = (e_lo + e_hi) >> 1; if (node_seg[mid] <= g) e_lo = mid + 1; else e_hi = mid; }

    float acc[B_DIM];
#pragma unroll
    for (int b = 0; b < B_DIM; ++b) acc[b] = 0.f;

    for (int i = s_lo + lane; i < e_lo; i += 32) {
        int   id = snp_ids[i];
        float w  = fsum[id];
#pragma unroll
        for (int b = 0; b < B_DIM; ++b) acc[b] += snp[b * N_SNPS + id] * w;
    }

#pragma unroll
    for (int b = 0; b < B_DIM; ++b)
#pragma unroll
        for (int m = 16; m >= 1; m >>= 1)
            acc[b] += __shfl_xor(acc[b], m, 32);

    if (lane == 0) {
#pragma unroll
        for (int b = 0; b < B_DIM; ++b) Apad[b * N_GENES + g] = acc[b] * (1.0f / N_FILT);
#pragma unroll
        for (int b = B_DIM; b < 16; ++b) Apad[b * N_GENES + g] = 0.f;
    }
}

// ---------------------------------------------------------------------------
// Kernel 3a: GEMM1 partial tiles via V_WMMA_F32_16X16X4_F32.
// grid = (ntiles=64, ksplit=25); block = 256 (8 waves, each K=100).
// Each block emits one partial 16x16 tile (after deterministic LDS reduce).
// ---------------------------------------------------------------------------
__global__ void k_gemm1_partial(const float* __restrict__ A,   // 16 x N_GENES
                                const float* __restrict__ W,   // N_GENES x H1
                                float*       __restrict__ part) {
    __shared__ float red[8 * 256];
    const int wave = threadIdx.x >> 5;
    const int lane = threadIdx.x & 31;
    const int n0   = blockIdx.x * 16;
    const int m    = lane & 15;
    const int kb   = (lane >> 4) << 1;   // 0 or 2

    v8f c = {};
    const int k0 = blockIdx.y * G1_KCHUNK + wave * G1_KWAVE;
#pragma unroll 5
    for (int kk = 0; kk < G1_KWAVE; kk += 4) {
        int k = k0 + kk;
        v2f a;
        a.x = A[m * N_GENES + k + kb];
        a.y = A[m * N_GENES + k + kb + 1];
        v2f bf;
        bf.x = W[(k + kb)     * H1 + n0 + m];
        bf.y = W[(k + kb + 1) * H1 + n0 + m];
        c = __builtin_amdgcn_wmma_f32_16x16x4_f32(false, a, false, bf,
                                                  (short)0, c, false, false);
    }

#pragma unroll
    for (int r = 0; r < 8; ++r) red[wave * 256 + r * 32 + lane] = c[r];
    __syncthreads();

    // 256 threads: deterministic cross-wave reduce, write partial tile
    float s = 0.f;
#pragma unroll
    for (int w = 0; w < 8; ++w) s += red[w * 256 + threadIdx.x];
    part[(blockIdx.y * G1_NTILES + blockIdx.x) * 256 + threadIdx.x] = s;
}

// ---------------------------------------------------------------------------
// Kernel 3b: reduce the 25 K-partials, fused bias + frozen-BN + ReLU,
// write h1 padded to 16 rows (rows 8..15 = 0).
// ---------------------------------------------------------------------------
__global__ void k_gemm1_reduce(const float* __restrict__ part,
                               const float* __restrict__ bias,
                               const float* __restrict__ gamma,
                               const float* __restrict__ beta,
                               float*       __restrict__ Out) {
    int idx = blockIdx.x * blockDim.x + threadIdx.x;    // over 64*256
    if (idx >= G1_NTILES * 256) return;
    int tile = idx >> 8, e = idx & 255;
    float s = 0.f;
#pragma unroll 5
    for (int p = 0; p < G1_KSPLIT; ++p) s += part[p * (G1_NTILES * 256) + idx];
    int r  = e >> 5, l = e & 31;
    int mm = r + ((l >> 4) << 3);
    int nn = tile * 16 + (l & 15);
    const float rs = rsqrtf(1.0f + 1e-5f);
    float o = 0.f;
    if (mm < B_DIM) {
        float x = s + bias[nn];
        o = fmaxf(x * (gamma[nn] * rs) + beta[nn], 0.f);
    }
    Out[mm * H1 + nn] = o;
}

// ---------------------------------------------------------------------------
// Kernel 4: fused in-block GEMM (used for GEMM2: 16x1024 @ 1024x256).
// One block per 16-column tile; waves split K; LDS reduce; fused epilogue.
// ---------------------------------------------------------------------------
__global__ void k_gemm_bn_relu(const float* __restrict__ A,
                               const float* __restrict__ W,
                               const float* __restrict__ bias,
                               const float* __restrict__ gamma,
                               const float* __restrict__ beta,
                               float*       __restrict__ Out,
                               int K, int N, int kPerWave) {
    __shared__ float red[8 * 256];
    const int nwaves = blockDim.x >> 5;
    const int wave   = threadIdx.x >> 5;
    const int lane   = threadIdx.x & 31;
    const int n0     = blockIdx.x * 16;
    const int m      = lane & 15;
    const int kb     = (lane >> 4) << 1;

    v8f c = {};
    const int k0 = wave * kPerWave;
    const int k1 = k0 + kPerWave;
#pragma unroll 4
    for (int k = k0; k < k1; k += 4) {
        v2f a;
        a.x = A[m * K + k + kb];
        a.y = A[m * K + k + kb + 1];
        v2f bf;
        bf.x = W[(k + kb)     * N + n0 + m];
        bf.y = W[(k + kb + 1) * N + n0 + m];
        c = __builtin_amdgcn_wmma_f32_16x16x4_f32(false, a, false, bf,
                                                  (short)0, c, false, false);
    }

#pragma unroll
    for (int r = 0; r < 8; ++r) red[wave * 256 + r * 32 + lane] = c[r];
    __syncthreads();

    const float rs = rsqrtf(1.0f + 1e-5f);
    for (int e = threadIdx.x; e < 256; e += blockDim.x) {
        float s = 0.f;
        for (int w = 0; w < nwaves; ++w) s += red[w * 256 + e];
        int r  = e >> 5, l = e & 31;
        int mm = r + ((l >> 4) << 3);
        int nn = l & 15;
        float o = 0.f;
        if (mm < B_DIM) {
            float x = s + bias[n0 + nn];
            o = fmaxf(x * (gamma[n0 + nn] * rs) + beta[n0 + nn], 0.f);
        }
        Out[mm * N + n0 + nn] = o;   // rows 8..15 explicitly zero-padded
    }
}

// ---------------------------------------------------------------------------
// Kernel 5: final 256->16 GEMM (single wave WMMA) + fused scalar head.
// ---------------------------------------------------------------------------
__global__ void k_gemm3_head(const float* __restrict__ A,   // 16 x 256 (padded)
                             const float* __restrict__ W3,
                             const float* __restrict__ b3,
                             const float* __restrict__ g3,
                             const float* __restrict__ be3,
                             const float* __restrict__ Wh1,
                             const float* __restrict__ bh1,
                             const float* __restrict__ gh,
                             const float* __restrict__ beh,
                             const float* __restrict__ Wh2,
                             const float* __restrict__ bh2,
                             float*       __restrict__ out) {
    __shared__ float feat[16 * 16];
    const int lane = threadIdx.x & 31;
    const int m    = lane & 15;
    const int kb   = (lane >> 4) << 1;

    v8f c = {};
#pragma unroll 4
    for (int k = 0; k < H2; k += 4) {
        v2f a;
        a.x = A[m * H2 + k + kb];
        a.y = A[m * H2 + k + kb + 1];
        v2f bf;
        bf.x = W3[(k + kb)     * H3 + m];
        bf.y = W3[(k + kb + 1) * H3 + m];
        c = __builtin_amdgcn_wmma_f32_16x16x4_f32(false, a, false, bf,
                                                  (short)0, c, false, false);
    }

    const float rs = rsqrtf(1.0f + 1e-5f);
#pragma unroll
    for (int r = 0; r < 8; ++r) {
        int mm = r + ((lane >> 4) << 3);
        int nn = lane & 15;
        float o = 0.f;
        if (mm < B_DIM) {
            float x = c[r] + b3[nn];
            o = fmaxf(x * (g3[nn] * rs) + be3[nn], 0.f);
        }
        feat[mm * 16 + nn] = o;
    }
    __syncthreads();

    if (lane < B_DIM) {
        float logit = bh2[0];
#pragma unroll
        for (int j = 0; j < 4; ++j) {
            float mj = bh1[j];
#pragma unroll
            for (int i = 0; i < 15; ++i) mj += feat[lane * 16 + i] * Wh1[i * 4 + j];
            mj = fmaxf(mj * (gh[j] * rs) + beh[j], 0.f);
            logit += mj * Wh2[j];
        }
        out[lane] = logit;
    }
}

// ---------------------------------------------------------------------------
extern "C" void kernel_launch(void* const* d_in, const int* in_sizes, int n_in,
                              void* d_out, int out_size, void* d_ws, size_t ws_size,
                              hipStream_t stream) {
    const float* snp      = (const float*)d_in[0];
    const int*   snp_ids  = (const int*)  d_in[1];
    const int*   node_seg = (const int*)  d_in[2];
    const float* filters  = (const float*)d_in[3];
    const float* W1  = (const float*)d_in[4];
    const float* b1  = (const float*)d_in[5];
    const float* g1  = (const float*)d_in[6];
    const float* be1 = (const float*)d_in[7];
    const float* W2  = (const float*)d_in[8];
    const float* b2  = (const float*)d_in[9];
    const float* g2  = (const float*)d_in[10];
    const float* be2 = (const float*)d_in[11];
    const float* W3  = (const float*)d_in[12];
    const float* b3  = (const float*)d_in[13];
    const float* g3  = (const float*)d_in[14];
    const float* be3 = (const float*)d_in[15];
    const float* Wh1 = (const float*)d_in[16];
    const float* bh1 = (const float*)d_in[17];
    const float* gh  = (const float*)d_in[18];
    const float* beh = (const float*)d_in[19];
    const float* Wh2 = (const float*)d_in[20];
    const float* bh2 = (const float*)d_in[21];

    float* ws   = (float*)d_ws;
    float* fsum = ws;                       // 500000 floats
    float* Apad = fsum + 500032;            // 16 x 20000
    float* h1   = Apad + 16 * N_GENES;      // 16 x 1024
    float* h2   = h1   + 16 * H1;           // 16 x 256
    float* part = h2   + 16 * H2;           // 25 x 64 x 256 = 409600 floats

    // 1) fold filters (16 MB stream)
    k_fsum<<<(N_SNPS + 255) / 256, 256, 0, stream>>>(filters, fsum);

    // 2) gather + segment-sum: one wave per gene (2500 blocks x 8 waves)
    k_segsum<<<N_GENES / 8, 256, 0, stream>>>(snp, snp_ids, node_seg, fsum, Apad);

    // 3) GEMM1: 64 n-tiles x 25 k-chunks = 1600 blocks streaming W1 once
    {
        dim3 grid(G1_NTILES, G1_KSPLIT);
        k_gemm1_partial<<<grid, 256, 0, stream>>>(Apad, W1, part);
        k_gemm1_reduce<<<(G1_NTILES * 256 + 255) / 256, 256, 0, stream>>>(
            part, b1, g1, be1, h1);
    }

    // 4) GEMM2: (16x1024) @ (1024x256), 16 n-tiles, 8 waves split K=1024
    k_gemm_bn_relu<<<H2 / 16, 256, 0, stream>>>(h1, W2, b2, g2, be2, h2,
                                                H1, H2, H1 / 8);

    // 5) GEMM3 + head: single wave
    k_gemm3_head<<<1, 32, 0, stream>>>(h2, W3, b3, g3, be3,
                                       Wh1, bh1, gh, beh, Wh2, bh2,
                                       (float*)d_out);
}